// SimpleAttentionLayer_48670569398981
// MI455X (gfx1250) — compile-verified
//
#include <hip/hip_runtime.h>

// Problem dims (compile-time constants from the reference)
#define B_  8
#define TE_ 256
#define TD_ 128
#define HE_ 512
#define HD_ 512

typedef __attribute__((ext_vector_type(2))) float v2f;
typedef __attribute__((ext_vector_type(8))) float v8f;

__device__ __forceinline__ float fast_tanh(float x) {
#if __has_builtin(__builtin_amdgcn_tanhf)
    return __builtin_amdgcn_tanhf(x);
#else
    float r;
    asm("v_tanh_f32 %0, %1" : "=v"(r) : "v"(x));
    return r;
#endif
}

// ---------------------------------------------------------------------------
// Kernel 1: projections via f32 WMMA.
//   blocks [0, 128)   : Ws = enc @ W_a, stored TRANSPOSED as WsT[b][h][t]
//   blocks [128, 192) : Uh = dec @ U_a, stored row-major Uh[b*TD+d][h]
// One block = one 16-row M-tile; 8 waves, each wave owns a 16x64 N-strip
// (4 accumulators). K = 512, stepped by 4 per V_WMMA_F32_16X16X4_F32.
// ---------------------------------------------------------------------------
__global__ __launch_bounds__(256) void proj_wmma(
    const float* __restrict__ enc, const float* __restrict__ dec,
    const float* __restrict__ W,   const float* __restrict__ U,
    float* __restrict__ wsT, float* __restrict__ uhm)
{
    const int lane   = threadIdx.x & 31;
    const int wave   = threadIdx.x >> 5;
    const bool is_dec = (blockIdx.x >= (B_ * TE_ / 16));
    const int mtile  = is_dec ? (blockIdx.x - B_ * TE_ / 16) : blockIdx.x;
    const int m0     = mtile * 16;

    const float* __restrict__ A  = is_dec ? dec : enc;   // [rows, 512]
    const float* __restrict__ Wt = is_dec ? U   : W;     // [512, 512]

    const int row     = m0 + (lane & 15);     // A-matrix M per lane
    const int kk_off  = 2 * (lane >> 4);      // K sub-offset per half-wave
    const int col_lo  = lane & 15;            // B/C/D N per lane
    const int n_base  = wave * 64;

    v8f acc[4] = {};

    for (int k = 0; k < HE_; k += 4) {
        const int kk = k + kk_off;
        // A tile 16x4: lane<16 -> {A[m][k],A[m][k+1]}, lane>=16 -> {A[m][k+2],A[m][k+3]}
        const float2 a2 = *(const float2*)(A + (size_t)row * HE_ + kk);
        v2f a; a.x = a2.x; a.y = a2.y;
#pragma unroll
        for (int j = 0; j < 4; ++j) {
            const int col = n_base + j * 16 + col_lo;
            // B tile 4x16: VGPR0 = W[kk][n], VGPR1 = W[kk+1][n]
            v2f bv;
            bv.x = Wt[(size_t)kk       * HE_ + col];
            bv.y = Wt[(size_t)(kk + 1) * HE_ + col];
            acc[j] = __builtin_amdgcn_wmma_f32_16x16x4_f32(
                false, a, false, bv, (short)0, acc[j], false, false);
        }
    }

    // C/D layout: element (r, lane) -> M = r + 8*(lane>>4), N = lane&15
    const int mrow = m0 + (lane >> 4) * 8;
    if (!is_dec) {
        const int bb = m0 / TE_;                       // 16-row tile never crosses b
        float* __restrict__ dst = wsT + (size_t)bb * HE_ * TE_;
#pragma unroll
        for (int j = 0; j < 4; ++j) {
            const int gc = n_base + j * 16 + col_lo;   // h
#pragma unroll
            for (int r = 0; r < 8; ++r) {
                const int t = (mrow + r) - bb * TE_;   // t
                dst[(size_t)gc * TE_ + t] = acc[j][r]; // WsT[b][h][t]
            }
        }
    } else {
#pragma unroll
        for (int j = 0; j < 4; ++j) {
            const int gc = n_base + j * 16 + col_lo;
#pragma unroll
            for (int r = 0; r < 8; ++r) {
                uhm[(size_t)(mrow + r) * HE_ + gc] = acc[j][r];
            }
        }
    }
}

// ---------------------------------------------------------------------------
// Kernel 2: fused logits + softmax. One block per (b, d); thread t owns
// encoder timestep t. 134M hardware v_tanh_f32 total across the grid.
//   e[t] = sum_h V[h] * tanh(WsT[b][h][t] + Uh[b,d][h])   (coalesced over t)
// then block-wide softmax over t via LDS tree reductions.
// ---------------------------------------------------------------------------
__global__ __launch_bounds__(256) void logits_softmax(
    const float* __restrict__ wsT, const float* __restrict__ uhm,
    const float* __restrict__ Va,  float* __restrict__ e_out)
{
    const int bd = blockIdx.x;          // b*TD + d
    const int b  = bd / TD_;
    const int t  = threadIdx.x;         // 0..255

    const float* __restrict__ wt = wsT + (size_t)b  * HE_ * TE_;
    const float* __restrict__ uh = uhm + (size_t)bd * HE_;

    float acc = 0.f;
#pragma unroll 8
    for (int h = 0; h < HE_; ++h) {
        const float s = wt[(size_t)h * TE_ + t] + uh[h];   // uh[h], Va[h] uniform
        acc = fmaf(Va[h], fast_tanh(s), acc);
    }

    __shared__ float red[256];
    red[t] = acc;
    __syncthreads();
#pragma unroll
    for (int s = 128; s > 0; s >>= 1) {
        if (t < s) red[t] = fmaxf(red[t], red[t + s]);
        __syncthreads();
    }
    const float mx = red[0];
    __syncthreads();
    const float ex = __expf(acc - mx);
    red[t] = ex;
    __syncthreads();
#pragma unroll
    for (int s = 128; s > 0; s >>= 1) {
        if (t < s) red[t] = red[t] + red[t + s];
        __syncthreads();
    }
    e_out[(size_t)bd * TE_ + t] = ex * (1.0f / red[0]);
}

// ---------------------------------------------------------------------------
// Kernel 3: contexts c[b] = e[b] @ enc[b]  ([128,256] x [256,512]) via f32
// WMMA. One block per (b, 16-row M-tile); 8 waves x 16x64 N-strips.
// ---------------------------------------------------------------------------
__global__ __launch_bounds__(256) void ctx_wmma(
    const float* __restrict__ e, const float* __restrict__ enc,
    float* __restrict__ cout)
{
    const int lane  = threadIdx.x & 31;
    const int wave  = threadIdx.x >> 5;
    const int b     = blockIdx.x >> 3;          // TD/16 = 8 m-tiles per batch
    const int mtile = blockIdx.x & 7;
    const int m0    = mtile * 16;

    const float* __restrict__ Ab = e   + (size_t)b * TD_ * TE_;   // [128,256]
    const float* __restrict__ Bb = enc + (size_t)b * TE_ * HE_;   // [256,512]

    const int row    = m0 + (lane & 15);
    const int kk_off = 2 * (lane >> 4);
    const int col_lo = lane & 15;
    const int n_base = wave * 64;

    v8f acc[4] = {};

    for (int k = 0; k < TE_; k += 4) {
        const int kk = k + kk_off;
        const float2 a2 = *(const float2*)(Ab + (size_t)row * TE_ + kk);
        v2f a; a.x = a2.x; a.y = a2.y;
#pragma unroll
        for (int j = 0; j < 4; ++j) {
            const int col = n_base + j * 16 + col_lo;
            v2f bv;
            bv.x = Bb[(size_t)kk       * HE_ + col];
            bv.y = Bb[(size_t)(kk + 1) * HE_ + col];
            acc[j] = __builtin_amdgcn_wmma_f32_16x16x4_f32(
                false, a, false, bv, (short)0, acc[j], false, false);
        }
    }

    const int mrow = m0 + (lane >> 4) * 8;
#pragma unroll
    for (int j = 0; j < 4; ++j) {
        const int gc = n_base + j * 16 + col_lo;
#pragma unroll
        for (int r = 0; r < 8; ++r) {
            cout[((size_t)b * TD_ + (mrow + r)) * HE_ + gc] = acc[j][r];
        }
    }
}

// ---------------------------------------------------------------------------
extern "C" void kernel_launch(void* const* d_in, const int* in_sizes, int n_in,
                              void* d_out, int out_size, void* d_ws, size_t ws_size,
                              hipStream_t stream) {
    const float* enc = (const float*)d_in[0];   // [8,256,512]
    const float* dec = (const float*)d_in[1];   // [8,128,512]
    const float* W   = (const float*)d_in[2];   // [512,512]
    const float* U   = (const float*)d_in[3];   // [512,512]
    const float* V   = (const float*)d_in[4];   // [512,1]

    float* c_out = (float*)d_out;                          // [8,128,512]
    float* e_out = c_out + (size_t)B_ * TD_ * HE_;         // [8,128,256]

    float* wsT = (float*)d_ws;                             // [8][512][256] = 4MB
    float* uhm = wsT + (size_t)B_ * HE_ * TE_;             // [8*128][512]  = 2MB

    // Stage 1: both projections (WMMA f32), Ws stored transposed.
    proj_wmma<<<(B_ * TE_ / 16) + (B_ * TD_ / 16), 256, 0, stream>>>(
        enc, dec, W, U, wsT, uhm);

    // Stage 2: fused tanh-logits + softmax (hardware v_tanh_f32).
    logits_softmax<<<B_ * TD_, 256, 0, stream>>>(wsT, uhm, V, e_out);

    // Stage 3: contexts (WMMA f32).
    ctx_wmma<<<B_ * (TD_ / 16), 256, 0, stream>>>(e_out, enc, c_out);
}